// LGNInputLayerCell_34170759807367
// MI455X (gfx1250) — compile-verified
//
#include <hip/hip_runtime.h>
#include <stdint.h>

// Problem constants from the reference
#define N_SOURCE 17400
#define N_TARGET 51978
#define N_SYN    20000000
#define MASK_WORDS ((N_SOURCE + 31) / 32)   // 544 words = 2176 B in LDS

typedef __attribute__((ext_vector_type(4))) int   v4i;
typedef __attribute__((ext_vector_type(4))) float v4f;

// Fire-and-forget FP32 global atomic add (no return -> STOREcnt, no CAS loop).
// Device scope so partial sums from all WGPs merge at L2. No "memory" clobber:
// the kernel never reads d_out, so the compiler can pipeline the streaming
// loads across these atomics; s_endpgm's implicit wait-idle drains STOREcnt.
__device__ __forceinline__ void gatomic_add_f32(float* p, float v) {
    asm volatile("global_atomic_add_f32 %0, %1, off scope:SCOPE_DEV"
                 :: "v"(p), "v"(v));
}

__device__ __forceinline__ void synapse(const unsigned* mask, float* out,
                                        int post, int pre, float val) {
    unsigned p = (unsigned)pre;
    if ((mask[p >> 5] >> (p & 31u)) & 1u)
        gatomic_add_f32(out + post, val);
}

// Main kernel: stream 320MB of synapse data with NT b128 loads, test the
// activity bit in LDS, scatter w*wf into out[post] with HW f32 atomics.
// 8 synapses per thread per iteration = 8 x global_load_b128 in flight.
__global__ __launch_bounds__(256) void scatter_kernel(const float* __restrict__ inputs_t,
                                                      const int*   __restrict__ indices,
                                                      const float* __restrict__ weights,
                                                      const float* __restrict__ factors,
                                                      const unsigned* __restrict__ gmask,
                                                      float* __restrict__ out) {
    __shared__ unsigned mask[MASK_WORDS];

    if (gmask != nullptr) {
        // Cheap path: 544 dword loads per block from the pre-packed bitmask.
        // Keep this compact -- it's a one-time 2.2KB copy, not worth unrolling.
        #pragma clang loop unroll(disable) vectorize(disable)
        for (int i = threadIdx.x; i < MASK_WORDS; i += blockDim.x)
            mask[i] = gmask[i];
    } else {
        // Fallback: build the bitmask from the float activity vector.
        #pragma clang loop unroll(disable) vectorize(disable)
        for (int i = threadIdx.x; i < MASK_WORDS; i += blockDim.x) {
            unsigned bits = 0u;
            int base = i * 32;
            int lim = N_SOURCE - base; if (lim > 32) lim = 32;
            #pragma clang loop unroll(disable) vectorize(disable)
            for (int b = 0; b < lim; ++b)
                if (inputs_t[base + b] > 0.0f) bits |= (1u << b);
            mask[i] = bits;
        }
    }
    __syncthreads();

    const v4i* idx4 = (const v4i*)indices;   // 4 x v4i per 8 synapses: [p,s,p,s]...
    const v4f* w4   = (const v4f*)weights;
    const v4f* f4   = (const v4f*)factors;

    const unsigned nGroups = N_SYN / 8u;     // 2,500,000 groups of 8 synapses
    const unsigned stride  = gridDim.x * blockDim.x;

    for (unsigned g = blockIdx.x * blockDim.x + threadIdx.x; g < nGroups; g += stride) {
        // Non-temporal: the 320MB single-pass stream must not thrash the
        // 192MB L2 holding the hot 208KB accumulator + bitmask.
        v4i i0 = __builtin_nontemporal_load(idx4 + 4u * g);
        v4i i1 = __builtin_nontemporal_load(idx4 + 4u * g + 1u);
        v4i i2 = __builtin_nontemporal_load(idx4 + 4u * g + 2u);
        v4i i3 = __builtin_nontemporal_load(idx4 + 4u * g + 3u);
        v4f w0 = __builtin_nontemporal_load(w4 + 2u * g);
        v4f w1 = __builtin_nontemporal_load(w4 + 2u * g + 1u);
        v4f f0 = __builtin_nontemporal_load(f4 + 2u * g);
        v4f f1 = __builtin_nontemporal_load(f4 + 2u * g + 1u);

        synapse(mask, out, i0.x, i0.y, w0.x * f0.x);
        synapse(mask, out, i0.z, i0.w, w0.y * f0.y);
        synapse(mask, out, i1.x, i1.y, w0.z * f0.z);
        synapse(mask, out, i1.z, i1.w, w0.w * f0.w);
        synapse(mask, out, i2.x, i2.y, w1.x * f1.x);
        synapse(mask, out, i2.z, i2.w, w1.y * f1.y);
        synapse(mask, out, i3.x, i3.y, w1.z * f1.z);
        synapse(mask, out, i3.z, i3.w, w1.w * f1.w);
    }
}

// Init kernel: zero the output accumulator (atomics accumulate and the harness
// replays without re-poisoning) and pack the activity into a global bitmask.
__global__ __launch_bounds__(256) void init_kernel(const float* __restrict__ inputs_t,
                                                   float* __restrict__ out,
                                                   unsigned* __restrict__ gmask) {
    int i = blockIdx.x * blockDim.x + threadIdx.x;
    if (i < N_TARGET) out[i] = 0.0f;
    if (gmask != nullptr && i < MASK_WORDS) {
        unsigned bits = 0u;
        int base = i * 32;
        int lim = N_SOURCE - base; if (lim > 32) lim = 32;
        #pragma clang loop unroll(disable) vectorize(disable)
        for (int b = 0; b < lim; ++b)
            if (inputs_t[base + b] > 0.0f) bits |= (1u << b);
        gmask[i] = bits;
    }
}

extern "C" void kernel_launch(void* const* d_in, const int* in_sizes, int n_in,
                              void* d_out, int out_size, void* d_ws, size_t ws_size,
                              hipStream_t stream) {
    // setup_inputs() order: inputs_t (f32), indices (i32), weights (f32), weights_factors (f32)
    const float* inputs_t = (const float*)d_in[0];
    const int*   indices  = (const int*)  d_in[1];
    const float* weights  = (const float*)d_in[2];
    const float* factors  = (const float*)d_in[3];
    float*       out      = (float*)d_out;

    unsigned* gmask = (ws_size >= MASK_WORDS * sizeof(unsigned)) ? (unsigned*)d_ws : nullptr;

    const int threads = 256;
    const int initBlocks = (N_TARGET + threads - 1) / threads;   // covers MASK_WORDS too
    init_kernel<<<initBlocks, threads, 0, stream>>>(inputs_t, out, gmask);

    // 2048 blocks x 8 wave32 = 16384 waves; each thread handles ~4.8 groups
    // of 8 synapses (8 outstanding b128 loads each) to saturate 23.3 TB/s.
    scatter_kernel<<<2048, threads, 0, stream>>>(inputs_t, indices, weights, factors, gmask, out);
}